// Nms_23450521436288
// MI455X (gfx1250) — compile-verified
//
#include <hip/hip_runtime.h>
#include <stdint.h>

// ---------------------------------------------------------------------------
// Greedy IoU NMS for MI455X (gfx1250).
//   B=64 batches, N=32768 boxes, K=300 kept, IoU threshold 0.5.
// One 1024-thread workgroup (32 wave32 waves = one full WGP) per batch.
//   - scores staged global->LDS via CDNA5 async DMA (global_load_async_to_lds),
//     then live in 32 VGPRs/thread and are mutated in place
//   - per-round argmax via 64-bit key (ordered score | ~idx) so unsigned max
//     reproduces jnp.argmax first-occurrence tie-breaking exactly
//   - wave32 butterfly (__shfl_xor) -> 32 LDS slots -> serial fold;
//     double-buffered slots => exactly ONE s_barrier per round
//   - suppression pass fused with next-round local argmax
// ---------------------------------------------------------------------------

#define NMS_B 64
#define NMS_N 32768
#define NMS_K 300
#define NMS_T 1024           // threads per block (32 waves of 32)
#define NMS_PER (NMS_N / NMS_T)   // 32 boxes per thread
#define NMS_IOU 0.5f
#define NMS_NEG (-1e30f)
#define NMS_I32MAX 2147483647

// total-order mapping: larger unsigned <=> larger float
__device__ __forceinline__ unsigned ordkey(float f) {
  unsigned u = __float_as_uint(f);
  return (u & 0x80000000u) ? ~u : (u | 0x80000000u);
}
__device__ __forceinline__ float ordinv(unsigned u) {
  return __uint_as_float((u & 0x80000000u) ? (u ^ 0x80000000u) : ~u);
}

extern "C" __global__ void __launch_bounds__(NMS_T)
nms_gfx1250_kernel(const float* __restrict__ scores,
                   const float* __restrict__ boxes,
                   const int*   __restrict__ classes,
                   float*       __restrict__ out)
{
  __shared__ float s_scores[NMS_N];                 // 128 KB of the 320 KB LDS
  __shared__ unsigned long long s_wmax[2][32];      // double-buffered wave maxima

  const int b    = blockIdx.x;
  const int tid  = threadIdx.x;
  const int wave = tid >> 5;
  const int lane = tid & 31;

  const float* gsc = scores + (size_t)b * NMS_N;
  const float* gbx = boxes  + (size_t)b * NMS_N * 4;

  // ---- CDNA5 async DMA: stage this batch's scores into LDS ----------------
  {
    // low 32 bits of the flat address of an LDS object == wave-relative LDS
    // byte offset (ISA 10.2)
    unsigned lds0 = (unsigned)(uintptr_t)&s_scores[0];
    unsigned long long gb = (unsigned long long)(uintptr_t)gsc;
    #pragma unroll
    for (int r = 0; r < (NMS_N * 4) / (NMS_T * 16); ++r) {   // 8 x b128 rounds
      unsigned byteoff = (unsigned)(tid + r * NMS_T) * 16u;
      unsigned ldsa = lds0 + byteoff;
      unsigned long long ga = gb + (unsigned long long)byteoff;
      asm volatile("global_load_async_to_lds_b128 %0, %1, off"
                   :: "v"(ldsa), "v"(ga) : "memory");
    }
    asm volatile("s_wait_asynccnt 0" ::: "memory");   // ASYNCcnt == 0
  }
  __syncthreads();

  // scores -> registers (each thread owns elements tid, tid+1024, ...)
  float sc[NMS_PER];
  #pragma unroll
  for (int j = 0; j < NMS_PER; ++j) sc[j] = s_scores[tid + (j << 10)];

  // initial local argmax key
  unsigned long long key = 0ull;
  #pragma unroll
  for (int j = 0; j < NMS_PER; ++j) {
    unsigned idx = (unsigned)(tid + (j << 10));
    unsigned long long k =
        ((unsigned long long)ordkey(sc[j]) << 32) | (unsigned)(~idx);
    key = (k > key) ? k : key;
  }

  int keep = 0;

  for (int it = 0; it < NMS_K; ++it) {
    // ---- wave32 butterfly max (5 shuffle steps) ----
    unsigned long long wk = key;
    #pragma unroll
    for (int m = 16; m >= 1; m >>= 1) {
      unsigned long long o = __shfl_xor(wk, m, 32);
      wk = (o > wk) ? o : wk;
    }
    if (lane == 0) s_wmax[it & 1][wave] = wk;
    __syncthreads();   // the only barrier per round (slots double-buffered)

    // every thread folds the 32 wave maxima (broadcast LDS reads)
    unsigned long long fk = s_wmax[it & 1][0];
    #pragma unroll
    for (int w = 1; w < 32; ++w) {
      unsigned long long o = s_wmax[it & 1][w];
      fk = (o > fk) ? o : fk;
    }

    const unsigned selIdx = ~(unsigned)fk;            // low 32 bits were ~idx
    const float    sval   = ordinv((unsigned)(fk >> 32));
    const bool     valid  = sval > (NMS_NEG * 0.5f);  // matches s[idx] > NEG/2

    // broadcast load of the selected box (uniform address -> single L2 req)
    const float4 sb = ((const float4*)gbx)[selIdx];
    const float sArea = (sb.z - sb.x) * (sb.w - sb.y);

    if (tid == 0) {
      const int base = b * NMS_K + it;
      int*   out_idx = (int*)out;                         // [B,K] int32
      float* out_sc  = out + NMS_B * NMS_K;               // [B,K] f32
      float* out_bx  = out + 2 * NMS_B * NMS_K;           // [B,K,4] f32
      int*   out_cl  = (int*)(out + 6 * NMS_B * NMS_K);   // [B,K] int32
      if (valid) {
        out_idx[base]      = (int)selIdx;
        out_sc[base]       = sval;
        out_bx[base * 4 + 0] = sb.x;
        out_bx[base * 4 + 1] = sb.y;
        out_bx[base * 4 + 2] = sb.z;
        out_bx[base * 4 + 3] = sb.w;
        out_cl[base]       = classes[(size_t)b * NMS_N + selIdx];
        keep++;
      } else {
        out_idx[base] = -1;
        out_sc[base]  = 0.0f;
        out_bx[base * 4 + 0] = 0.0f;
        out_bx[base * 4 + 1] = 0.0f;
        out_bx[base * 4 + 2] = 0.0f;
        out_bx[base * 4 + 3] = 0.0f;
        out_cl[base]  = NMS_I32MAX;
      }
    }

    // ---- fused suppression + next-round local argmax ----
    key = 0ull;
    #pragma unroll
    for (int j = 0; j < NMS_PER; ++j) {
      const unsigned idx = (unsigned)(tid + (j << 10));
      const float4 mb = ((const float4*)gbx)[idx];   // coalesced, L2-resident
      float iw = fminf(mb.z, sb.z) - fmaxf(mb.x, sb.x);
      float ih = fminf(mb.w, sb.w) - fmaxf(mb.y, sb.y);
      iw = fmaxf(iw, 0.0f);
      ih = fmaxf(ih, 0.0f);
      const float inter = iw * ih;
      const float area  = (mb.z - mb.x) * (mb.w - mb.y);
      const float iou   = inter / (area + sArea - inter);  // exact, matches ref
      if (valid && (iou > NMS_IOU)) sc[j] = NMS_NEG;
      unsigned long long k2 =
          ((unsigned long long)ordkey(sc[j]) << 32) | (unsigned)(~idx);
      key = (k2 > key) ? k2 : key;
    }
  }

  if (tid == 0) ((int*)(out + 7 * NMS_B * NMS_K))[b] = keep;  // true_max[B]
}

// ---------------------------------------------------------------------------
extern "C" void kernel_launch(void* const* d_in, const int* in_sizes, int n_in,
                              void* d_out, int out_size, void* d_ws, size_t ws_size,
                              hipStream_t stream) {
  (void)in_sizes; (void)n_in; (void)d_ws; (void)ws_size; (void)out_size;
  const float* scores  = (const float*)d_in[0];   // [B,N] f32
  const float* boxes   = (const float*)d_in[1];   // [B,N,4] f32
  const int*   classes = (const int*)d_in[2];     // [B,N] i32
  nms_gfx1250_kernel<<<NMS_B, NMS_T, 0, stream>>>(scores, boxes, classes,
                                                  (float*)d_out);
}